// OperatorProduct_18932215841386
// MI455X (gfx1250) — compile-verified
//
#include <hip/hip_runtime.h>

// ---------------------------------------------------------------------------
// exp(-i*t*H) @ psi  via scaling-and-squaring Taylor on U, then one big apply.
// All matrix math runs on V_WMMA_F32_16X16X4_F32 (fp32 WMMA, exact precision).
// Working set (X + ping/pong U = 6 planes x 9.4MB = 57MB) is L2-resident
// (192MB L2), so the D x D GEMM chain is compute-bound on the fp32 WMMA pipe.
// ---------------------------------------------------------------------------

typedef __attribute__((ext_vector_type(2))) float v2f;
typedef __attribute__((ext_vector_type(8))) float v8f;

#define DD   1536
#define NN2  ((size_t)DD * (size_t)DD)
#define BB   4096

// ---------------------------------------------------------------------------
// X = (-i * t / 2^s) * Hermitize(Hre + i*Him)
//   Xre =  scale * (Him - Him^T)/2
//   Xim = -scale * (Hre + Hre^T)/2
// ---------------------------------------------------------------------------
__global__ __launch_bounds__(256)
void build_X_kernel(const float* __restrict__ Hre, const float* __restrict__ Him,
                    float* __restrict__ Xre, float* __restrict__ Xim, float scale) {
    size_t idx = (size_t)blockIdx.x * blockDim.x + threadIdx.x;
    if (idx >= NN2) return;
    int i = (int)(idx / DD);
    int j = (int)(idx % DD);
    float hr = 0.5f * (Hre[(size_t)i * DD + j] + Hre[(size_t)j * DD + i]);
    float hi = 0.5f * (Him[(size_t)i * DD + j] - Him[(size_t)j * DD + i]);
    Xre[idx] =  scale * hi;
    Xim[idx] = -scale * hr;
}

__global__ __launch_bounds__(256)
void set_identity_kernel(float* __restrict__ Pre, float* __restrict__ Pim) {
    size_t idx = (size_t)blockIdx.x * blockDim.x + threadIdx.x;
    if (idx >= NN2) return;
    int i = (int)(idx / DD);
    int j = (int)(idx % DD);
    Pre[idx] = (i == j) ? 1.0f : 0.0f;
    Pim[idx] = 0.0f;
}

// ---------------------------------------------------------------------------
// Complex GEMM:  C = alpha * (A @ B) [+ I]      (planar re/im storage)
//   A: DD x DD   (row-major, lda = DD)
//   B: DD x N    (row-major, ldb = N)
//   C: DD x N
// MODE 0: write planar Cre/Cim (optional +I on the real part)
// MODE 1: write interleaved complex64 into Cre (Cim unused) -- final output
//
// Block: 256 threads = 8 waves, arranged 2(M) x 4(N); block tile 64 x 256.
// Wave tile: 32 x 64 = 2x4 grid of 16x16 WMMA tiles, complex ->
//   16 v8f accumulators (128 VGPRs).
// Inner step: k += 4, 32 x v_wmma_f32_16x16x4_f32 against 4 b64 + 16 b32 loads:
//   accRe += Ar*Br;  accRe += Ai*(-Bi);  accIm += Ar*Bi;  accIm += Ai*Br;
// (f32 WMMA NEG bits only cover the C operand, so Bi is negated with VALU.)
// ---------------------------------------------------------------------------
template <int MODE>
__global__ __launch_bounds__(256)
void zgemm_wmma_kernel(const float* __restrict__ Are, const float* __restrict__ Aim,
                       const float* __restrict__ Bre, const float* __restrict__ Bim,
                       float* __restrict__ Cre, float* __restrict__ Cim,
                       int N, float alpha, int addI) {
    const int lane  = threadIdx.x & 31;
    const int wave  = threadIdx.x >> 5;
    const int waveM = wave & 1;   // 2 waves along M
    const int waveN = wave >> 1;  // 4 waves along N

    const int m0 = blockIdx.y * 64  + waveM * 32;
    const int n0 = blockIdx.x * 256 + waveN * 64;

    const int lr = lane & 15;   // row (A) / col (B,C) within 16-tile
    const int kh = lane >> 4;   // K-half for A/B frags, M-half for C tiles
    const int kb = kh * 2;      // K sub-offset within the k-quad

    v8f accRe[2][4] = {};
    v8f accIm[2][4] = {};

    const size_t ldb = (size_t)N;

    for (int k = 0; k < DD; k += 4) {
        v2f ar[2], ai[2], br[4], bi[4], bin[4];

        // A fragments: lane holds A[m, k+kb] , A[m, k+kb+1]  (contiguous -> b64)
#pragma unroll
        for (int mt = 0; mt < 2; ++mt) {
            const size_t rowOff = (size_t)(m0 + mt * 16 + lr) * DD + (size_t)(k + kb);
            ar[mt] = *(const v2f*)(Are + rowOff);
            ai[mt] = *(const v2f*)(Aim + rowOff);
        }
        // B fragments: lane holds B[k+kb, n] , B[k+kb+1, n]  (stride ldb)
#pragma unroll
        for (int nt = 0; nt < 4; ++nt) {
            const size_t off = (size_t)(k + kb) * ldb + (size_t)(n0 + nt * 16 + lr);
            br[nt].x = Bre[off];
            br[nt].y = Bre[off + ldb];
            bi[nt].x = Bim[off];
            bi[nt].y = Bim[off + ldb];
            bin[nt]  = -bi[nt];
        }

#pragma unroll
        for (int mt = 0; mt < 2; ++mt) {
#pragma unroll
            for (int nt = 0; nt < 4; ++nt) {
                accRe[mt][nt] = __builtin_amdgcn_wmma_f32_16x16x4_f32(
                    false, ar[mt], false, br[nt], (short)0, accRe[mt][nt], false, false);
                accRe[mt][nt] = __builtin_amdgcn_wmma_f32_16x16x4_f32(
                    false, ai[mt], false, bin[nt], (short)0, accRe[mt][nt], false, false);
                accIm[mt][nt] = __builtin_amdgcn_wmma_f32_16x16x4_f32(
                    false, ar[mt], false, bi[nt], (short)0, accIm[mt][nt], false, false);
                accIm[mt][nt] = __builtin_amdgcn_wmma_f32_16x16x4_f32(
                    false, ai[mt], false, br[nt], (short)0, accIm[mt][nt], false, false);
            }
        }
    }

    // Epilogue. C/D layout: lane = (kh,lr): col = lr(+tile), VGPR r -> row r + 8*kh.
#pragma unroll
    for (int mt = 0; mt < 2; ++mt) {
#pragma unroll
        for (int nt = 0; nt < 4; ++nt) {
            const int col = n0 + nt * 16 + lr;
#pragma unroll
            for (int r = 0; r < 8; ++r) {
                const int row = m0 + mt * 16 + kh * 8 + r;
                float re = alpha * accRe[mt][nt][r];
                float im = alpha * accIm[mt][nt][r];
                if (MODE == 0) {
                    if (addI && row == col) re += 1.0f;
                    Cre[(size_t)row * ldb + col] = re;
                    Cim[(size_t)row * ldb + col] = im;
                } else {
                    // interleaved complex64 output
                    v2f o; o.x = re; o.y = im;
                    *(v2f*)(Cre + 2u * ((size_t)row * ldb + col)) = o;
                }
            }
        }
    }
}

// ---------------------------------------------------------------------------
extern "C" void kernel_launch(void* const* d_in, const int* in_sizes, int n_in,
                              void* d_out, int out_size, void* d_ws, size_t ws_size,
                              hipStream_t stream) {
    const float* Hre    = (const float*)d_in[0];
    const float* Him    = (const float*)d_in[1];
    const float* psi_re = (const float*)d_in[2];
    const float* psi_im = (const float*)d_in[3];

    float* ws = (float*)d_ws;
    float* Xre = ws;
    float* Xim = ws + NN2;
    float* Ure[2] = { ws + 2 * NN2, ws + 4 * NN2 };
    float* Uim[2] = { ws + 3 * NN2, ws + 5 * NN2 };

    const int   S     = 6;                       // squarings: ||t*H||/2^S ~ 0.12
    const int   K     = 12;                      // Taylor order at scaled norm
    const float scale = 0.1f / (float)(1 << S);  // t / 2^S

    {
        const int thr = 256;
        const int blks = (int)((NN2 + thr - 1) / thr);
        build_X_kernel<<<blks, thr, 0, stream>>>(Hre, Him, Xre, Xim, scale);
        set_identity_kernel<<<blks, thr, 0, stream>>>(Ure[0], Uim[0]);
    }

    dim3 blk(256);
    dim3 grdD(DD / 256, DD / 64);   // 6 x 24 blocks for D x D GEMMs
    int cur = 0;

    // Horner:  P <- I + (X @ P) / k ,  k = K .. 1
    for (int k = K; k >= 1; --k) {
        zgemm_wmma_kernel<0><<<grdD, blk, 0, stream>>>(
            Xre, Xim, Ure[cur], Uim[cur], Ure[1 - cur], Uim[1 - cur],
            DD, 1.0f / (float)k, 1);
        cur = 1 - cur;
    }
    // Squaring:  U <- U @ U , S times
    for (int i = 0; i < S; ++i) {
        zgemm_wmma_kernel<0><<<grdD, blk, 0, stream>>>(
            Ure[cur], Uim[cur], Ure[cur], Uim[cur], Ure[1 - cur], Uim[1 - cur],
            DD, 1.0f, 0);
        cur = 1 - cur;
    }
    // Apply:  out = U @ psi  (interleaved complex64 into d_out)
    dim3 grdB(BB / 256, DD / 64);   // 16 x 24 blocks
    zgemm_wmma_kernel<1><<<grdB, blk, 0, stream>>>(
        Ure[cur], Uim[cur], psi_re, psi_im, (float*)d_out, nullptr,
        BB, 1.0f, 0);
}